// ANIRepresentation_5677946766113
// MI455X (gfx1250) — compile-verified
//
#include <hip/hip_runtime.h>
#include <cmath>
#include <cstdint>

#define BLK 256
#define PI_D 3.14159265358979323846

typedef float f4 __attribute__((ext_vector_type(4)));

// CDNA5 async global->LDS path (gfx1250). Round-2 histogram proved this path
// lowers (ds ops + async wait present; the fallback has no LDS at all).
#if defined(__HIP_DEVICE_COMPILE__) && defined(__gfx1250__) &&                  \
    __has_builtin(__builtin_amdgcn_global_load_async_to_lds_b32) &&             \
    __has_builtin(__builtin_amdgcn_s_wait_asynccnt)
#define USE_ASYNC_LDS 1
#else
#define USE_ASYNC_LDS 0
#endif

__device__ __forceinline__ void store_nt4(float* p, float a, float b, float c, float d) {
    f4 v = {a, b, c, d};
    __builtin_nontemporal_store(v, (f4*)p);
}

// ---------------- radial: rfv [P,16] + scatter into radial_aev [N*7,16] -----
__global__ __launch_bounds__(BLK) void ani_radial(
    const float* __restrict__ d_ij,
    const int*   __restrict__ atomic_numbers,
    const int*   __restrict__ pair_indices,   // [2,P]
    float*       __restrict__ rfv_out,        // [P,16]
    float*       __restrict__ aev_out,        // [N*7,16], pre-zeroed
    int P)
{
    int p = blockIdx.x * BLK + threadIdx.x;
    if (p >= P) return;

    // stream-ahead prefetch of d_ij (global_prefetch_b8, confirmed in asm)
    __builtin_prefetch(d_ij + ((p + 16 * BLK < P) ? (p + 16 * BLK) : p), 0, 0);

    // single-use streaming load: non-temporal, keep L2 for the reuse sets
    float d = __builtin_nontemporal_load(d_ij + p);
    const float pir = (float)(PI_D / 0.51);          // pi / RCUT
    float fc = (d <= 0.51f) ? (0.5f * __cosf(pir * d) + 0.5f) : 0.0f;
    float coef = 0.25f * fc;

    float v[16];
#pragma unroll
    for (int s = 0; s < 16; ++s) {
        const float shfr = (float)(0.08 + s * ((0.51 - 0.08) / 16.0));
        float x = d - shfr;
        v[s] = coef * __expf(-1970.0f * x * x);
    }

    float* o = rfv_out + (size_t)p * 16;
    store_nt4(o + 0,  v[0],  v[1],  v[2],  v[3]);
    store_nt4(o + 4,  v[4],  v[5],  v[6],  v[7]);
    store_nt4(o + 8,  v[8],  v[9],  v[10], v[11]);
    store_nt4(o + 12, v[12], v[13], v[14], v[15]);

    // index12[0] = i*7 + species[j]; index12[1] = j*7 + species[i]
    int ai = __builtin_nontemporal_load(pair_indices + p);
    int aj = __builtin_nontemporal_load(pair_indices + P + p);
    int zi = atomic_numbers[ai];
    int zj = atomic_numbers[aj];
    float* r0 = aev_out + ((size_t)ai * 7 + zj) * 16;
    float* r1 = aev_out + ((size_t)aj * 7 + zi) * 16;
#pragma unroll
    for (int s = 0; s < 16; ++s) {
        atomicAdd(r0 + s, v[s]);   // global_atomic_add_f32 into the L2-resident 9MB accumulator
        atomicAdd(r1 + s, v[s]);
    }
}

// ---------------- angular: ang [T,32] over triples --------------------------
__global__ __launch_bounds__(BLK) void ani_angular(
    const float* __restrict__ r_ij,          // [P,3]
    const int*   __restrict__ close_idx,     // [Pc]
    const int*   __restrict__ pair_index12,  // [2,T]
    const int*   __restrict__ sign12,        // [2,T]
    float*       __restrict__ ang_out,       // [T,32]
    int T)
{
    int tid = threadIdx.x;
    int t   = blockIdx.x * BLK + tid;
    int tc  = (t < T) ? t : (T - 1);         // clamp so all lanes issue valid loads

    int p0, p1;
    float sg0, sg1;
#if USE_ASYNC_LDS
    __shared__ int s_p0[BLK], s_p1[BLK], s_s0[BLK], s_s1[BLK];
    __builtin_amdgcn_global_load_async_to_lds_b32((int*)(pair_index12 + tc),     &s_p0[tid], 0, 0);
    __builtin_amdgcn_global_load_async_to_lds_b32((int*)(pair_index12 + T + tc), &s_p1[tid], 0, 0);
    __builtin_amdgcn_global_load_async_to_lds_b32((int*)(sign12 + tc),           &s_s0[tid], 0, 0);
    __builtin_amdgcn_global_load_async_to_lds_b32((int*)(sign12 + T + tc),       &s_s1[tid], 0, 0);
    __builtin_amdgcn_s_wait_asynccnt(0);
    // each lane consumes only its own staged slot -> no barrier needed
    p0 = s_p0[tid]; p1 = s_p1[tid];
    sg0 = (float)s_s0[tid]; sg1 = (float)s_s1[tid];
#else
    p0 = pair_index12[tc];  p1 = pair_index12[T + tc];
    sg0 = (float)sign12[tc]; sg1 = (float)sign12[T + tc];
#endif
    if (t >= T) return;

    // double-indirect gather of the two pair displacement vectors (reused
    // ~14x across triples -> default temporal policy, lives in L2)
    int c0 = close_idx[p0];
    int c1 = close_idx[p1];
    const float* a = r_ij + (size_t)c0 * 3;
    const float* b = r_ij + (size_t)c1 * 3;
    float x0 = a[0] * sg0, y0 = a[1] * sg0, z0 = a[2] * sg0;
    float x1 = b[0] * sg1, y1 = b[1] * sg1, z1 = b[2] * sg1;

    float d0 = sqrtf(x0 * x0 + y0 * y0 + z0 * z0);
    float d1 = sqrtf(x1 * x1 + y1 * y1 + z1 * z1);
    float ct = 0.95f * (x0 * x1 + y0 * y1 + z0 * z1) / (d0 * d1);
    float st = sqrtf(fmaxf(1.0f - ct * ct, 0.0f));   // sin(arccos(ct)) >= 0

    const float pia = (float)(PI_D / 0.35);          // pi / ACUT
    float fc0 = (d0 <= 0.35f) ? (0.5f * __cosf(pia * d0) + 0.5f) : 0.0f;
    float fc1 = (d1 <= 0.35f) ? (0.5f * __cosf(pia * d1) + 0.5f) : 0.0f;
    float pref = 2.0f * fc0 * fc1;

    float mean = 0.5f * (d0 + d1);
    float f2[8];
#pragma unroll
    for (int k = 0; k < 8; ++k) {
        const float shfa = (float)(0.08 + k * ((0.35 - 0.08) / 8.0));
        float x = mean - shfa;
        f2[k] = __expf(-800.0f * x * x);
    }

    // SHFZ = pi/8 + z*pi/4 ; f1 = ((1 + cos(theta - SHFZ))/2)^32
    // cos(theta - SHFZ) = ct*cos(SHFZ) + st*sin(SHFZ); ^32 via 5 squarings.
    const float CZ[4] = { 0.92387953251128674f,  0.38268343236508978f,
                         -0.38268343236508978f, -0.92387953251128674f};
    const float SZ[4] = { 0.38268343236508978f,  0.92387953251128674f,
                          0.92387953251128674f,  0.38268343236508978f};

    // dominant ~290MB write-once stream: non-temporal b128 stores
    float* o = ang_out + (size_t)t * 32;
#pragma unroll
    for (int z = 0; z < 4; ++z) {
        float base = 0.5f * (1.0f + ct * CZ[z] + st * SZ[z]);
        float q = base * base;  // ^2
        q *= q;                 // ^4
        q *= q;                 // ^8
        q *= q;                 // ^16
        q *= q;                 // ^32
        float g = pref * q;
        store_nt4(o + 8 * z,     g * f2[0], g * f2[1], g * f2[2], g * f2[3]);
        store_nt4(o + 8 * z + 4, g * f2[4], g * f2[5], g * f2[6], g * f2[7]);
    }
}

// ---------------- launch ----------------------------------------------------
extern "C" void kernel_launch(void* const* d_in, const int* in_sizes, int n_in,
                              void* d_out, int out_size, void* d_ws, size_t ws_size,
                              hipStream_t stream)
{
    const float* d_ij = (const float*)d_in[0];
    const float* r_ij = (const float*)d_in[1];
    const int*   atn  = (const int*)d_in[2];
    const int*   pidx = (const int*)d_in[3];
    const int*   cidx = (const int*)d_in[4];
    const int*   p12  = (const int*)d_in[5];
    const int*   s12  = (const int*)d_in[6];

    const int P = in_sizes[0];       // pairs
    const int N = in_sizes[2];       // atoms
    const int T = in_sizes[5] / 2;   // triples

    float* rfv = (float*)d_out;                    // [P,16]
    float* ang = rfv + (size_t)P * 16;             // [T,32]
    float* aev = ang + (size_t)T * 32;             // [N*7,16]

    // zero the scatter accumulator (capture-safe memset node)
    (void)hipMemsetAsync(aev, 0, (size_t)N * 7 * 16 * sizeof(float), stream);

    if (P > 0) {
        dim3 g((P + BLK - 1) / BLK);
        ani_radial<<<g, BLK, 0, stream>>>(d_ij, atn, pidx, rfv, aev, P);
    }
    if (T > 0) {
        dim3 g((T + BLK - 1) / BLK);
        ani_angular<<<g, BLK, 0, stream>>>(r_ij, cidx, p12, s12, ang, T);
    }
}